// Aggregation_74904229642960
// MI455X (gfx1250) — compile-verified
//
#include <hip/hip_runtime.h>
#include <stdint.h>

#define N_NODES   10000
#define N_EDGES   640000
#define DFEAT     128
#define D4        (DFEAT / 4)     // 32 float4 per row -> one wave32 per edge row
#define EPB       8               // 8 waves (256 threads) per block = 8 edges/block

typedef float v4f __attribute__((ext_vector_type(4)));

// ---- CDNA5 native float atomics, device scope, no-return (STOREcnt) ----
__device__ __forceinline__ void atom_max_f32_dev(float* p, float v) {
  asm volatile("global_atomic_max_num_f32 %0, %1, off scope:SCOPE_DEV"
               :: "v"(p), "v"(v) : "memory");
}
__device__ __forceinline__ void atom_add_f32_dev(float* p, float v) {
  asm volatile("global_atomic_add_f32 %0, %1, off scope:SCOPE_DEV"
               :: "v"(p), "v"(v) : "memory");
}

__device__ __forceinline__ void wait_async0() {
#if __has_builtin(__builtin_amdgcn_s_wait_asynccnt)
  __builtin_amdgcn_s_wait_asynccnt(0);
#else
  asm volatile("s_wait_asynccnt 0" ::: "memory");
#endif
}

// ---- split async index staging: issue early, finish late ----------------
// Lane 0 of each wave issues a 4B global->LDS async copy of this wave's
// target-node index; the wave's 512B x-row load is issued NEXT (before the
// ASYNCcnt wait) so both memory latencies overlap. finish() waits and
// broadcasts the index from the wave-private LDS slot (no barrier needed).
__device__ __forceinline__ void stage_tgt_issue(const int* __restrict__ tgt_arr,
                                                int e, int lane, int wv, int* ltgt) {
  if (lane == 0) {
    uint32_t lds_off = (uint32_t)(uintptr_t)(&ltgt[wv]);  // low 32b of generic = LDS offset
    const int* gp = tgt_arr + e;
    asm volatile("global_load_async_to_lds_b32 %0, %1, off"
                 :: "v"(lds_off), "v"(gp) : "memory");
  }
}
__device__ __forceinline__ int stage_tgt_finish(int wv, const int* ltgt) {
  wait_async0();
  return ltgt[wv];
}

// x streams 327MB/pass (exceeds 192MB L2) -> non-temporal keeps the node
// accumulators (15MB) resident in L2 across all three passes.
__device__ __forceinline__ v4f load_x_row_nt(const float* __restrict__ x,
                                             int e, int lane) {
  return __builtin_nontemporal_load(
      (const v4f*)(x + (size_t)e * DFEAT + lane * 4));
}

// ---- init: max = -inf, sum = 0, out = 0 ----
__global__ __launch_bounds__(256) void init_kernel(float* __restrict__ m,
                                                   float* __restrict__ s,
                                                   float* __restrict__ out) {
  int i = blockIdx.x * blockDim.x + threadIdx.x;
  if (i < N_NODES * D4) {
    const float ninf = -__builtin_inff();
    v4f vninf = {ninf, ninf, ninf, ninf};
    v4f vz    = {0.f, 0.f, 0.f, 0.f};
    ((v4f*)m)[i]   = vninf;
    ((v4f*)s)[i]   = vz;
    ((v4f*)out)[i] = vz;
  }
}

// ---- pass 1: segment max via native f32 max atomics ----
__global__ __launch_bounds__(256) void pass1_max(const float* __restrict__ x,
                                                 const int* __restrict__ ei,
                                                 float* __restrict__ m) {
  __shared__ int ltgt[EPB];
  const int lane = threadIdx.x & 31;
  const int wv   = threadIdx.x >> 5;
  const int e    = blockIdx.x * EPB + wv;

  stage_tgt_issue(ei + N_EDGES, e, lane, wv, ltgt);   // row 1 of [2,E]
  const v4f xv  = load_x_row_nt(x, e, lane);          // in flight with index fetch
  const int tgt = stage_tgt_finish(wv, ltgt);

  float* mp = m + (size_t)tgt * DFEAT + lane * 4;
  atom_max_f32_dev(mp + 0, xv.x);
  atom_max_f32_dev(mp + 1, xv.y);
  atom_max_f32_dev(mp + 2, xv.z);
  atom_max_f32_dev(mp + 3, xv.w);
}

// ---- pass 2: e = exp(x - max[tgt]); segment sum via native f32 add atomics ----
__global__ __launch_bounds__(256) void pass2_expsum(const float* __restrict__ x,
                                                    const int* __restrict__ ei,
                                                    const float* __restrict__ m,
                                                    float* __restrict__ s) {
  __shared__ int ltgt[EPB];
  const int lane = threadIdx.x & 31;
  const int wv   = threadIdx.x >> 5;
  const int e    = blockIdx.x * EPB + wv;

  stage_tgt_issue(ei + N_EDGES, e, lane, wv, ltgt);
  const v4f xv  = load_x_row_nt(x, e, lane);
  const int tgt = stage_tgt_finish(wv, ltgt);

  const v4f mv = *(const v4f*)(m + (size_t)tgt * DFEAT + lane * 4);  // L2-hot gather

  v4f ev;
  ev.x = __expf(xv.x - mv.x);
  ev.y = __expf(xv.y - mv.y);
  ev.z = __expf(xv.z - mv.z);
  ev.w = __expf(xv.w - mv.w);

  float* sp = s + (size_t)tgt * DFEAT + lane * 4;
  atom_add_f32_dev(sp + 0, ev.x);
  atom_add_f32_dev(sp + 1, ev.y);
  atom_add_f32_dev(sp + 2, ev.z);
  atom_add_f32_dev(sp + 3, ev.w);
}

// ---- pass 3: out[tgt] += exp(x - max[tgt]) / sum[tgt] ----
__global__ __launch_bounds__(256) void pass3_scatter(const float* __restrict__ x,
                                                     const int* __restrict__ ei,
                                                     const float* __restrict__ m,
                                                     const float* __restrict__ s,
                                                     float* __restrict__ out) {
  __shared__ int ltgt[EPB];
  const int lane = threadIdx.x & 31;
  const int wv   = threadIdx.x >> 5;
  const int e    = blockIdx.x * EPB + wv;

  stage_tgt_issue(ei + N_EDGES, e, lane, wv, ltgt);
  const v4f xv  = load_x_row_nt(x, e, lane);
  const int tgt = stage_tgt_finish(wv, ltgt);

  const v4f mv = *(const v4f*)(m + (size_t)tgt * DFEAT + lane * 4);
  const v4f sv = *(const v4f*)(s + (size_t)tgt * DFEAT + lane * 4);

  v4f r;
  r.x = __expf(xv.x - mv.x) / sv.x;
  r.y = __expf(xv.y - mv.y) / sv.y;
  r.z = __expf(xv.z - mv.z) / sv.z;
  r.w = __expf(xv.w - mv.w) / sv.w;

  float* op = out + (size_t)tgt * DFEAT + lane * 4;
  atom_add_f32_dev(op + 0, r.x);
  atom_add_f32_dev(op + 1, r.y);
  atom_add_f32_dev(op + 2, r.z);
  atom_add_f32_dev(op + 3, r.w);
}

extern "C" void kernel_launch(void* const* d_in, const int* in_sizes, int n_in,
                              void* d_out, int out_size, void* d_ws, size_t ws_size,
                              hipStream_t stream) {
  const float* x  = (const float*)d_in[0];   // [E, 128] fp32
  const int*   ei = (const int*)d_in[1];     // [2, E]; row 1 = targets
  float*       out = (float*)d_out;          // [N, 128] fp32

  // workspace: max[N,128] | sum[N,128]  (2 * 5.12 MB)
  float* m = (float*)d_ws;
  float* s = m + (size_t)N_NODES * DFEAT;

  (void)in_sizes; (void)n_in; (void)out_size; (void)ws_size;

  const int init_blocks = (N_NODES * D4 + 255) / 256;
  const int edge_blocks = N_EDGES / EPB;     // 80000, exact

  init_kernel  <<<init_blocks, 256, 0, stream>>>(m, s, out);
  pass1_max    <<<edge_blocks, 256, 0, stream>>>(x, ei, m);
  pass2_expsum <<<edge_blocks, 256, 0, stream>>>(x, ei, m, s);
  pass3_scatter<<<edge_blocks, 256, 0, stream>>>(x, ei, m, s, out);
}